// EntityModel_78125455114637
// MI455X (gfx1250) — compile-verified
//
#include <hip/hip_runtime.h>

typedef float v2f __attribute__((ext_vector_type(2)));
typedef float v8f __attribute__((ext_vector_type(8)));

#define NB 64
#define NS 512
#define NH 768
#define NTAG 17
#define NPOS 45
#define K4STEPS (NH / 4)   // 192
#define NTILES 5           // 2 for tag (pad 17->32), 3 for pos (pad 45->48)

// ---------------------------------------------------------------------------
// Kernel 0: pre-swizzle W_tag / W_pos into WMMA B-fragment order.
// bfrag layout: [k4][ntile][lane] of float2, i.e. lane l gets
//   b.x = W[(4*k4 + 2*(l>>4))  ][n],  b.y = W[(4*k4 + 2*(l>>4))+1][n]
// with n = ntile*16 + (l&15); out-of-range columns are zero-padded.
// ---------------------------------------------------------------------------
__global__ void build_bfrag(const float* __restrict__ Wt,
                            const float* __restrict__ Wp,
                            float* __restrict__ bfrag) {
    int idx = blockIdx.x * blockDim.x + threadIdx.x;   // 0 .. 192*5*32-1
    if (idx >= K4STEPS * NTILES * 32) return;
    int lane = idx & 31;
    int nt   = (idx >> 5) % NTILES;
    int k4   = idx / (NTILES * 32);
    int hi = lane >> 4, lo = lane & 15;
    int kk = k4 * 4 + 2 * hi;
    int n  = nt * 16 + lo;
    float x = 0.f, y = 0.f;
    if (nt < 2) {                      // tag head, cols 0..31 (valid < 17)
        if (n < NTAG) { x = Wt[kk * NTAG + n]; y = Wt[(kk + 1) * NTAG + n]; }
    } else {                           // pos head, cols 0..47 (valid < 45)
        int c = n - 32;
        if (c < NPOS) { x = Wp[kk * NPOS + c]; y = Wp[(kk + 1) * NPOS + c]; }
    }
    bfrag[idx * 2 + 0] = x;
    bfrag[idx * 2 + 1] = y;
}

// ---------------------------------------------------------------------------
// Kernel 1: fused emissions GEMM (both heads) via V_WMMA_F32_16X16X4_F32.
// One wave per 16-row tile of hidden; 5 accumulators (2 tag tiles + 3 pos
// tiles) so each hidden element is loaded exactly once. 192 K-steps.
// ---------------------------------------------------------------------------
__global__ void __launch_bounds__(32)
emis_gemm(const float* __restrict__ hidden,
          const float* __restrict__ bfrag,
          const float* __restrict__ b_tag,
          const float* __restrict__ b_pos,
          float* __restrict__ em_tag,    // [B*S, 17]
          float* __restrict__ em_pos) {  // [B*S, 45]
    const int tile = blockIdx.x;         // 0 .. 2047 (32768 rows / 16)
    const int lane = threadIdx.x;        // wave32
    const int hi = lane >> 4, lo = lane & 15;

    // A fragment addressing: M = lo, K-pair = 2*hi within each 4-wide K block
    const float* arow = hidden + ((size_t)tile * 16 + lo) * NH + 2 * hi;
    const float* bln  = bfrag + lane * 2;

    v8f c0 = {}, c1 = {}, c2 = {}, c3 = {}, c4 = {};
    for (int k4 = 0; k4 < K4STEPS; ++k4) {
        v2f a = *(const v2f*)(arow + k4 * 4);
        const float* bk = bln + k4 * (NTILES * 64);
        v2f b0 = *(const v2f*)(bk + 0 * 64);
        v2f b1 = *(const v2f*)(bk + 1 * 64);
        v2f b2 = *(const v2f*)(bk + 2 * 64);
        v2f b3 = *(const v2f*)(bk + 3 * 64);
        v2f b4 = *(const v2f*)(bk + 4 * 64);
        c0 = __builtin_amdgcn_wmma_f32_16x16x4_f32(false, a, false, b0, (short)0, c0, false, false);
        c1 = __builtin_amdgcn_wmma_f32_16x16x4_f32(false, a, false, b1, (short)0, c1, false, false);
        c2 = __builtin_amdgcn_wmma_f32_16x16x4_f32(false, a, false, b2, (short)0, c2, false, false);
        c3 = __builtin_amdgcn_wmma_f32_16x16x4_f32(false, a, false, b3, (short)0, c3, false, false);
        c4 = __builtin_amdgcn_wmma_f32_16x16x4_f32(false, a, false, b4, (short)0, c4, false, false);
    }

    // C/D layout: VGPR r -> row M = r + 8*hi, col N = lo (+ tile offset)
    for (int r = 0; r < 8; ++r) {
        size_t row = (size_t)tile * 16 + r + 8 * hi;
        int ct0 = lo;                                   // tag tile 0
        em_tag[row * NTAG + ct0] = c0[r] + b_tag[ct0];
        int ct1 = 16 + lo;                              // tag tile 1
        if (ct1 < NTAG) em_tag[row * NTAG + ct1] = c1[r] + b_tag[ct1];
        int cp0 = lo;                                   // pos tiles
        em_pos[row * NPOS + cp0] = c2[r] + b_pos[cp0];
        int cp1 = 16 + lo;
        em_pos[row * NPOS + cp1] = c3[r] + b_pos[cp1];
        int cp2 = 32 + lo;
        if (cp2 < NPOS) em_pos[row * NPOS + cp2] = c4[r] + b_pos[cp2];
    }
}

// ---------------------------------------------------------------------------
// Kernel 2: per-(batch,head) CRF — fused forward logsumexp (denominator),
// target-path numerator, Viterbi forward with LDS backpointers, and the
// backtrace. log_softmax is omitted: CRF NLL and Viterbi argmax are both
// invariant to per-(b,t) constant emission shifts.
// ---------------------------------------------------------------------------
template <int T>
__global__ void __launch_bounds__(64)
crf_chain(const float* __restrict__ em,      // [B*S, T]
          const int* __restrict__ mask,      // [B, S]
          const int* __restrict__ tgt,       // [B, S]
          const float* __restrict__ startv,  // [T]
          const float* __restrict__ transv,  // [T, T]
          const float* __restrict__ endv,    // [T]
          float* __restrict__ pred,          // [B, S] (written as float)
          float* __restrict__ loss_part) {   // [B]  (num - den)
    const int b = blockIdx.x;
    const int j = threadIdx.x;

    __shared__ float tr[T * T];
    __shared__ float sc[T];                  // forward (logsumexp) scores
    __shared__ float vs[T];                  // Viterbi scores
    __shared__ float red[T];
    __shared__ unsigned char hist[(NS - 1) * T];

    for (int i = j; i < T * T; i += blockDim.x) tr[i] = transv[i];

    const float* eb = em + (size_t)b * NS * T;
    const float NEG = -3.0e38f;

    float myS = 0.f, myV = 0.f;
    if (j < T) {
        float e0 = eb[j];
        myS = startv[j] + e0;
        myV = myS;
        sc[j] = myS;
        vs[j] = myV;
    }

    float num = 0.f;
    int   prev = 0;
    if (j == 0) {
        int t0 = tgt[b * NS];
        prev = t0;
        num  = startv[t0] + eb[t0];
    }
    __syncthreads();

    for (int t = 1; t < NS; ++t) {
        const float* et = eb + t * T;
        const int mt = mask[b * NS + t];
        float nS = myS, nV = myV;
        int bp = j;
        if (j < T) {
            float ej = et[j];
            float mx = NEG, bestv = NEG;
            int bi = 0;
#pragma unroll 4
            for (int i = 0; i < T; ++i) {
                float trv = tr[i * T + j];
                float a = sc[i] + trv;
                mx = fmaxf(mx, a);
                float c = vs[i] + trv;
                if (c > bestv) { bestv = c; bi = i; }
            }
            float sum = 0.f;
#pragma unroll 4
            for (int i = 0; i < T; ++i)
                sum += __expf(sc[i] + tr[i * T + j] - mx);
            if (mt > 0) {
                nS = mx + __logf(sum) + ej;
                nV = bestv + ej;
                bp = bi;
            }
            hist[(t - 1) * T + j] = (unsigned char)bp;
        }
        if (j == 0 && mt > 0) {
            int ct = tgt[b * NS + t];
            num += tr[prev * T + ct] + et[ct];
            prev = ct;
        }
        __syncthreads();           // all reads of sc/vs done
        if (j < T) { myS = nS; myV = nV; sc[j] = nS; vs[j] = nV; }
        __syncthreads();           // new scores visible
    }

    // denominator: logsumexp(score + end)
    if (j < T) red[j] = myS + endv[j];
    __syncthreads();
    if (j == 0) {
        float mx = NEG;
        for (int i = 0; i < T; ++i) mx = fmaxf(mx, red[i]);
        float s = 0.f;
        for (int i = 0; i < T; ++i) s += __expf(red[i] - mx);
        float den = mx + __logf(s);
        num += endv[prev];
        loss_part[b] = num - den;
    }
    __syncthreads();

    // Viterbi termination + backtrace (backpointers live in LDS)
    if (j < T) red[j] = myV + endv[j];
    __syncthreads();
    if (j == 0) {
        float best = NEG;
        int last = 0;
        for (int i = 0; i < T; ++i)
            if (red[i] > best) { best = red[i]; last = i; }
        int tg = last;
        for (int t = NS - 1; t >= 1; --t) {
            pred[b * NS + t] = (float)tg;
            tg = hist[(t - 1) * T + tg];
        }
        pred[b * NS] = (float)tg;
    }
}

// ---------------------------------------------------------------------------
// Kernel 3: final loss reduction  loss = -mean_b(num - den)
// ---------------------------------------------------------------------------
__global__ void reduce_loss(const float* __restrict__ part_tag,
                            const float* __restrict__ part_pos,
                            float* __restrict__ out) {
    if (threadIdx.x == 0 && blockIdx.x == 0) {
        float a = 0.f, p = 0.f;
        for (int i = 0; i < NB; ++i) { a += part_tag[i]; p += part_pos[i]; }
        out[2 * NB * NS + 0] = -a / (float)NB;
        out[2 * NB * NS + 1] = -p / (float)NB;
    }
}

// ---------------------------------------------------------------------------
extern "C" void kernel_launch(void* const* d_in, const int* in_sizes, int n_in,
                              void* d_out, int out_size, void* d_ws, size_t ws_size,
                              hipStream_t stream) {
    const float* hidden    = (const float*)d_in[0];
    const int*   mask      = (const int*)d_in[1];
    const int*   tgt_tag   = (const int*)d_in[2];
    const int*   tgt_pos   = (const int*)d_in[3];
    const float* W_tag     = (const float*)d_in[4];
    const float* b_tag     = (const float*)d_in[5];
    const float* W_pos     = (const float*)d_in[6];
    const float* b_pos     = (const float*)d_in[7];
    const float* start_tag = (const float*)d_in[8];
    const float* trans_tag = (const float*)d_in[9];
    const float* end_tag   = (const float*)d_in[10];
    const float* start_pos = (const float*)d_in[11];
    const float* trans_pos = (const float*)d_in[12];
    const float* end_pos   = (const float*)d_in[13];

    float* out = (float*)d_out;

    // workspace carving (sizes are 16B-aligned by construction)
    char* ws = (char*)d_ws;
    const size_t SZ_BFRAG = (size_t)K4STEPS * NTILES * 32 * 2 * sizeof(float); // 245760
    const size_t SZ_EMTAG = (size_t)NB * NS * NTAG * sizeof(float);            // 2228224
    const size_t SZ_EMPOS = (size_t)NB * NS * NPOS * sizeof(float);            // 5898240
    float* bfrag    = (float*)(ws);
    float* em_tag   = (float*)(ws + SZ_BFRAG);
    float* em_pos   = (float*)(ws + SZ_BFRAG + SZ_EMTAG);
    float* part_tag = (float*)(ws + SZ_BFRAG + SZ_EMTAG + SZ_EMPOS);
    float* part_pos = part_tag + NB;

    // 0) swizzle W into WMMA B-fragment order
    int nfrag = K4STEPS * NTILES * 32;
    build_bfrag<<<(nfrag + 255) / 256, 256, 0, stream>>>(W_tag, W_pos, bfrag);

    // 1) fused two-head emissions GEMM (f32 WMMA)
    emis_gemm<<<(NB * NS) / 16, 32, 0, stream>>>(hidden, bfrag, b_tag, b_pos,
                                                 em_tag, em_pos);

    // 2) CRF scans, one workgroup per (batch, head)
    crf_chain<NTAG><<<NB, 64, 0, stream>>>(em_tag, mask, tgt_tag,
                                           start_tag, trans_tag, end_tag,
                                           out, part_tag);
    crf_chain<NPOS><<<NB, 64, 0, stream>>>(em_pos, mask, tgt_pos,
                                           start_pos, trans_pos, end_pos,
                                           out + NB * NS, part_pos);

    // 3) losses
    reduce_loss<<<1, 32, 0, stream>>>(part_tag, part_pos, out);
}